// GNN_59365037965408
// MI455X (gfx1250) — compile-verified
//
#include <hip/hip_runtime.h>
#include <math.h>

typedef float v2f __attribute__((ext_vector_type(2)));
typedef float v8f __attribute__((ext_vector_type(8)));

#define B_BATCH   512
#define L_ROW     569      // 25 + 16*34
#define SEG_LEN_  59
#define N_ALLELE_ 16
#define K_CAND_   8
#define D_MODEL_  256
#define VOCAB_    30
#define NEG_FILL_ -1000.0f
#define NCHUNK    15       // ceil(59/4)
#define LUT_PITCH 61       // odd pitch -> gcd(61,64)=1, conflict-free LDS banks

// D = A x B with A = W_pool (M=16 candidates, 8 valid) and B = S^T (rows along N).
// C/D layout: VGPR j, lanes 0-15 -> M=j (candidate j), N=lane (row): every lane
// holds all 8 candidate preds for its row -> lane-local argmax, no shuffles.
// 8192 rows, 16 rows per wave -> 512 waves = 128 blocks x 128 threads.
__global__ __launch_bounds__(128) void mhc_wmma_kernel(
    const int*   __restrict__ x,      // (512, 569) int32 tokens
    const float* __restrict__ emb,    // (30, 256)
    const float* __restrict__ Wpool,  // (8, 59)
    const float* __restrict__ Wpred,  // (256)
    const float* __restrict__ bpred,  // (1)
    float*       __restrict__ out)    // (512, 32)
{
  __shared__ float e_dot[32];                  // e_dot[v] = dot(emb[v], Wpred)
  __shared__ float pe_dot[64];                 // pe_dot[l] = dot(pe[l], Wpred)
  __shared__ float lut[VOCAB_ * LUT_PITCH];    // lut[v][l]; row0 and cols>=59 are 0

  const int tid = threadIdx.x;

  // ---- phase 0a: 1D LUTs (the d=256 dimension collapses into these) ----
  if (tid < VOCAB_) {
    float s = 0.f;
    #pragma unroll 8
    for (int d = 0; d < D_MODEL_; ++d) s += emb[tid * D_MODEL_ + d] * Wpred[d];
    e_dot[tid] = s;
  }
  if (tid >= 64 && tid < 64 + SEG_LEN_) {
    const int l = tid - 64;
    const float pos = (float)l;
    const float c = -logf(10000.0f) / (float)D_MODEL_;
    float s = 0.f;
    for (int i = 0; i < D_MODEL_ / 2; ++i) {
      float ang = pos * expf((float)(2 * i) * c);
      s += sinf(ang) * Wpred[2 * i] + cosf(ang) * Wpred[2 * i + 1];
    }
    pe_dot[l] = s;
  }
  __syncthreads();

  // ---- phase 0b: fused 2D LUT; tok==0 mask and l>=59 padding baked in ----
  for (int i = tid; i < VOCAB_ * LUT_PITCH; i += 128) {
    const int v = i / LUT_PITCH;
    const int l = i - v * LUT_PITCH;
    lut[i] = (v == 0 || l >= SEG_LEN_) ? 0.f : (e_dot[v] + pe_dot[l]);
  }
  __syncthreads();

  const int lane = tid & 31;
  const int half = lane >> 4;   // 0: K pair {0,1}, 1: K pair {2,3}
  const int lidx = lane & 15;   // A: row M (candidate) ; B/C/D: col N (row n)

  // ---- A fragments: W_pool tile (16 x 4 per chunk), M>=8 zero-padded ----
  // All loads unconditional (clamped addresses); masking via value selects.
  const int wrow = (lidx < K_CAND_) ? lidx : 0;
  v2f afrag[NCHUNK];
  #pragma unroll
  for (int c = 0; c < NCHUNK; ++c) {
    const int l0 = 4 * c + 2 * half;       // <= 58 always
    const int l1 = l0 + 1;                 // == 59 only at c==14, half==1
    float w0 = Wpool[wrow * SEG_LEN_ + l0];
    float w1 = Wpool[wrow * SEG_LEN_ + ((l1 < SEG_LEN_) ? l1 : SEG_LEN_ - 1)];
    afrag[c].x = (lidx < K_CAND_) ? w0 : 0.f;
    afrag[c].y = (lidx < K_CAND_ && l1 < SEG_LEN_) ? w1 : 0.f;
  }

  // ---- per-wave row tile ----
  const int wave    = (int)(blockIdx.x * blockDim.x + tid) >> 5;
  const int rowBase = wave * 16;               // 16 consecutive n; same allele
  const int a       = rowBase >> 9;            // allele = n / 512 (wave-uniform)
  const int bb      = (rowBase & 511) + lidx;  // batch index of this lane's row
  const int xrow    = bb * L_ROW;
  const int mhcBase = 25 + 34 * a;

  // ---- stage 1: batch token gathers (unconditional, clamped columns) ----
  int tok0[NCHUNK], tok1[NCHUNK];
  #pragma unroll
  for (int c = 0; c < NCHUNK; ++c) {
    const int l0 = 4 * c + 2 * half;
    const int l1c = (l0 + 1 < SEG_LEN_) ? (l0 + 1) : (SEG_LEN_ - 1);  // clamp
    const int col0 = (l0  < 25) ? l0  : (mhcBase + l0  - 25);
    const int col1 = (l1c < 25) ? l1c : (mhcBase + l1c - 25);
    tok0[c] = x[xrow + col0];
    tok1[c] = x[xrow + col1];
  }

  // ---- stage 2: batch LUT lookups (one unconditional ds_load per element) ----
  // l==59 hits the zero-padded column regardless of token value.
  v2f bfrag[NCHUNK];
  #pragma unroll
  for (int c = 0; c < NCHUNK; ++c) {
    const int l0 = 4 * c + 2 * half;
    const int l1 = l0 + 1;
    bfrag[c].x = lut[tok0[c] * LUT_PITCH + l0];
    bfrag[c].y = lut[tok1[c] * LUT_PITCH + l1];
  }

  // ---- stage 3: 15 back-to-back f32 WMMAs (exact accumulation) ----
  v8f acc = {};
  #pragma unroll
  for (int c = 0; c < NCHUNK; ++c) {
    acc = __builtin_amdgcn_wmma_f32_16x16x4_f32(
        /*neg_a=*/false, afrag[c], /*neg_b=*/false, bfrag[c],
        /*c_mod=*/(short)0, acc, /*reuse_a=*/false, /*reuse_b=*/false);
  }

  // ---- lane-local max/argmax over candidates (acc[0..7] = k=0..7) ----
  float best = acc[0];
  int   barg = 0;
  #pragma unroll
  for (int j = 1; j < 8; ++j) {
    const bool t = acc[j] > best;      // strict > keeps first max (jnp.argmax)
    best = t ? acc[j] : best;
    barg = t ? j : barg;
  }

  // ---- writeback: lanes 0-15 (half==0) own rows; presence token = tok1[6] (l=25) ----
  if (half == 0) {
    const bool pres = (tok1[6] != 0);  // mhc[:,0] != 0 for this row
    out[bb * 32 + a]      = pres ? (best + bpred[0]) : NEG_FILL_;
    out[bb * 32 + 16 + a] = pres ? (float)barg : -1.0f;
  }
}

extern "C" void kernel_launch(void* const* d_in, const int* in_sizes, int n_in,
                              void* d_out, int out_size, void* d_ws, size_t ws_size,
                              hipStream_t stream) {
  const int*   x     = (const int*)d_in[0];    // x_data (512,569) int
  const float* emb   = (const float*)d_in[1];  // (30,256)
  const float* Wpool = (const float*)d_in[2];  // (8,59)
  const float* Wpred = (const float*)d_in[3];  // (256,)
  const float* bpred = (const float*)d_in[4];  // scalar
  float* out = (float*)d_out;                  // (512,32)

  (void)in_sizes; (void)n_in; (void)out_size; (void)d_ws; (void)ws_size;

  // 8192 rows / 16 rows-per-wave = 512 waves = 128 blocks x 4 waves
  mhc_wmma_kernel<<<128, 128, 0, stream>>>(x, emb, Wpool, Wpred, bpred, out);
}